// ChemConv_20521353740396
// MI455X (gfx1250) — compile-verified
//
#include <hip/hip_runtime.h>
#include <cstddef>

// Problem constants (from the reference)
#define NA   2048          // atoms
#define ND   128           // IN_DEPTH
#define NO   128           // OUT_DEPTH
#define NF   16            // FILTER_LEN
#define FROW (ND + 2)      // 130 floats per (o,f) filter row
#define KBIG (NA * NF)     // 32768 : flattened (b,f) contraction axis
#define KSPLIT 16          // K-split factor for the big GEMM
#define KCHUNK (KBIG / KSPLIT)   // 2048 k per block
#define KT     128               // k rows staged in LDS per stage
#define NSTAGE (KCHUNK / KT)     // 16 stages per block
#define LPITCH 136               // LDS row pitch (floats): 128 + 8 pad -> disjoint banks

typedef __attribute__((ext_vector_type(2))) float v2f;
typedef __attribute__((ext_vector_type(8))) float v8f;
typedef __attribute__((ext_vector_type(4))) int   v4i;

typedef __attribute__((address_space(1))) v4i g_v4i;   // global int4
typedef __attribute__((address_space(3))) v4i l_v4i;   // LDS int4

// D = A(16x4, f32) * B(4x16, f32) + C(16x16, f32)  -- wave32 collaborative
__device__ __forceinline__ v8f wmma_f32_16x16x4(v2f a, v2f b, v8f c) {
    return __builtin_amdgcn_wmma_f32_16x16x4_f32(
        /*neg_a=*/false, a, /*neg_b=*/false, b,
        /*c_mod=*/(short)0, c, /*reuse_a=*/false, /*reuse_b=*/false);
}

// ---- async global -> LDS staging (CDNA5 GLOBAL_LOAD_ASYNC_TO_LDS_B128) ----
#if defined(__has_builtin)
#if __has_builtin(__builtin_amdgcn_global_load_async_to_lds_b128)
#define HAVE_ASYNC_LDS 1
#endif
#endif
#ifndef HAVE_ASYNC_LDS
#define HAVE_ASYNC_LDS 0
#endif

__device__ __forceinline__ void copy_b128_to_lds(const float* g, float* l) {
#if HAVE_ASYNC_LDS
    __builtin_amdgcn_global_load_async_to_lds_b128(
        (g_v4i*)g, (l_v4i*)l, /*imm offset=*/0, /*cpol=*/0);
#else
    *(float4*)l = *(const float4*)g;    // sync fallback: global_load_b128 + ds_store_b128
#endif
}

__device__ __forceinline__ void stage_fence() {
#if HAVE_ASYNC_LDS
    asm volatile("s_wait_asynccnt 0x0" ::: "memory");  // own async copies landed in LDS
#endif
    __syncthreads();                                   // all waves' copies visible
}

// ---------------------------------------------------------------------------
// Kernel 1: G[(b*16+f), o] = sum_d node[b,d] * filters[o,f,d]
// One wave computes a 16(b) x 16(o) tile for a fixed f, K=128 over d.
// grid = (NA/16, NF), block = 256 threads (8 waves; wave w -> o-tile w).
// ---------------------------------------------------------------------------
__global__ void chemconv_make_g(const float* __restrict__ node,
                                const float* __restrict__ filters,
                                float* __restrict__ G) {
    const int lane = threadIdx.x & 31;
    const int wave = threadIdx.x >> 5;      // o tile: 0..7
    const int mn   = lane & 15;             // M row (A frag) / N col (B frag)
    const int kk   = (lane >> 4) * 2;       // lanes 16-31 hold K+2,K+3 (ISA 7.12.2)
    const int b0   = blockIdx.x * 16;
    const int f    = blockIdx.y;
    const int o0   = wave * 16;

    v8f acc = {};
    const float* arow = node + (size_t)(b0 + mn) * ND + kk;
    const float* bcol = filters + (size_t)(o0 + mn) * (NF * FROW)
                                + (size_t)f * FROW + kk;
    #pragma unroll 4
    for (int k0 = 0; k0 < ND; k0 += 4) {
        v2f a = *(const v2f*)(arow + k0);
        v2f b;
        b.x = bcol[k0];
        b.y = bcol[k0 + 1];
        acc = wmma_f32_16x16x4(a, b, acc);
    }
    #pragma unroll
    for (int i = 0; i < 8; ++i) {
        const int m = (lane < 16) ? i : (i + 8);
        G[(size_t)((b0 + m) * NF + f) * NO + o0 + (lane & 15)] = acc[i];
    }
}

// ---------------------------------------------------------------------------
// Kernel 2: out[a,o] = sum_{f,j<2} bond[a,f,j] * filters[o,f,128+j]
// Tiny; also initializes d_out before the atomic GEMM reduction.
// ---------------------------------------------------------------------------
__global__ void chemconv_bias_init(const float* __restrict__ bond,
                                   const float* __restrict__ filters,
                                   float* __restrict__ out) {
    const int idx = blockIdx.x * blockDim.x + threadIdx.x;   // a*NO + o
    if (idx >= NA * NO) return;
    const int a = idx >> 7;
    const int o = idx & (NO - 1);
    const float* bp = bond + (size_t)a * (NF * 2);
    const float* fp = filters + (size_t)o * (NF * FROW) + ND;
    float s = 0.f;
    #pragma unroll
    for (int f = 0; f < NF; ++f) {
        s += bp[2 * f]     * fp[f * FROW];
        s += bp[2 * f + 1] * fp[f * FROW + 1];
    }
    out[idx] = s;
}

// ---------------------------------------------------------------------------
// Kernel 3: C(2048x128) += conn(2048x32768) * G(32768x128)
//  - conn streamed once from HBM with b64 loads (23.3 TB/s roofline floor)
//  - G staged through LDS with double-buffered async b128 copies (ASYNCcnt),
//    B fragments served by ds_load -> one VMEM stream, one LDS stream, WMMA.
//  - block = 8 waves on one 16-row M stripe (A dedup in WGP$); wave w owns
//    N-tile w; grid.y splits K 16x; f32 atomic reduction into out.
// ---------------------------------------------------------------------------
__global__ void __launch_bounds__(256)
chemconv_gemm(const float* __restrict__ conn,
              const float* __restrict__ G,
              float* __restrict__ out) {
    __shared__ float gbuf[2][KT * LPITCH];   // 2 x 68 KB

    const int t    = threadIdx.x;
    const int lane = t & 31;
    const int wave = t >> 5;                 // N tile: 0..7
    const int mn   = lane & 15;
    const int kk   = (lane >> 4) * 2;
    const int a0   = blockIdx.x * 16;
    const int kb   = blockIdx.y * KCHUNK;
    const int o0   = wave * 16;

    // Stage copier: KT rows x 8 b128 segs = 1024 chunks, 4 per thread.
    auto issue_stage = [&](int s, int buf) {
        const float* gsrc = G + (size_t)(kb + s * KT) * NO;
        float* ldst = &gbuf[buf][0];
        #pragma unroll
        for (int i = 0; i < 4; ++i) {
            const int c   = t + i * 256;
            const int row = c >> 3;
            const int seg = (c & 7) * 4;
            copy_b128_to_lds(gsrc + (size_t)row * NO + seg,
                             ldst + row * LPITCH + seg);
        }
    };

    v8f acc = {};
    issue_stage(0, 0);
    for (int s = 0; s < NSTAGE; ++s) {
        stage_fence();                                  // buf[s&1] ready for all waves
        if (s + 1 < NSTAGE) issue_stage(s + 1, (s + 1) & 1);   // prefetch next buffer

        const float* arow = conn + (size_t)(a0 + mn) * KBIG + kb + s * KT + kk;
        const float* brow = &gbuf[s & 1][kk * LPITCH + o0 + mn];
        #pragma unroll 8
        for (int k0 = 0; k0 < KT; k0 += 4) {
            v2f a = *(const v2f*)(arow + k0);           // global_load_b64 (HBM stream)
            v2f b;
            b.x = brow[(k0    ) * LPITCH];              // ds_load_b32
            b.y = brow[(k0 + 1) * LPITCH];
            acc = wmma_f32_16x16x4(a, b, acc);
        }
    }

    #pragma unroll
    for (int i = 0; i < 8; ++i) {
        const int m = (lane < 16) ? i : (i + 8);
        atomicAdd(&out[(size_t)(a0 + m) * NO + o0 + (lane & 15)], acc[i]);
    }
}

// ---------------------------------------------------------------------------
extern "C" void kernel_launch(void* const* d_in, const int* in_sizes, int n_in,
                              void* d_out, int out_size, void* d_ws, size_t ws_size,
                              hipStream_t stream) {
    const float* node = (const float*)d_in[0];   // (A, D)
    const float* conn = (const float*)d_in[1];   // (A, A, F)
    const float* bond = (const float*)d_in[2];   // (A, F, 2)
    const float* filt = (const float*)d_in[3];   // (O, F, D+2)
    float* out = (float*)d_out;                  // (A, O)
    float* G   = (float*)d_ws;                   // (KBIG, NO) = 16.8 MB scratch

    chemconv_make_g<<<dim3(NA / 16, NF), 256, 0, stream>>>(node, filt, G);
    chemconv_bias_init<<<(NA * NO) / 256, 256, 0, stream>>>(bond, filt, out);
    chemconv_gemm<<<dim3(NA / 16, KSPLIT), 256, 0, stream>>>(conn, G, out);
}